// JVSLoss_695784702578
// MI455X (gfx1250) — compile-verified
//
#include <hip/hip_runtime.h>
#include <hip/hip_bf16.h>

typedef float v2f __attribute__((ext_vector_type(2)));
typedef float v8f __attribute__((ext_vector_type(8)));

static __device__ __forceinline__ v8f wmma_f32(v2f a, v2f b, v8f c) {
    // D = A(16x4) x B(4x16) + C, fp32 throughout
    return __builtin_amdgcn_wmma_f32_16x16x4_f32(false, a, false, b, (short)0, c,
                                                 false, false);
}

namespace {
constexpr int kB = 16;
constexpr int kC = 256;
constexpr int kT = 1024;

constexpr int TILE_T = 128;     // rows (t) per workgroup
constexpr int TILE_S = 64;      // cols (s) per workgroup
constexpr int KC     = 16;      // K (channel) chunk staged per LDS buffer
constexpr int KP     = KC / 2;  // K pairs per buffer
constexpr int NCHUNK = kC / KC; // 16

// LDS tiles hold float2 {K=2kp, K=2kp+1} per column. Fragment reads are
// ds_load_b64; lanes 0-15 read row kp, lanes 16-31 read row kp+1. For the two
// half-wave 32-bank windows to be disjoint: 2*stride % 64 == 32 (float2 units).
constexpr int LD2T = 144;       // stride (float2) for 128-wide tiles
constexpr int LD2S = 80;        // stride (float2) for 64-wide tiles
}  // namespace

__global__ void jvs_zero(float* __restrict__ out) {
    if (threadIdx.x == 0) out[0] = 0.0f;
}

struct F4x4 { float4 x0, x1, y0, y1; };  // t-unit prefetch (x and y tiles)
struct F4x2 { float4 a0, a1; };          // s-unit prefetch (x or y tile)

__global__ __launch_bounds__(256) void jvs_main(const float* __restrict__ inp,
                                                const float* __restrict__ tgt,
                                                const float* __restrict__ mask,
                                                float* __restrict__ out) {
    // Double-buffered, pair-interleaved tiles.
    __shared__ float sXT[2][KP * LD2T * 2];   // x, cols = t
    __shared__ float sYT[2][KP * LD2T * 2];   // y, cols = t
    __shared__ float sXS[2][KP * LD2S * 2];   // x, cols = s
    __shared__ float sYS[2][KP * LD2S * 2];   // y, cols = s
    __shared__ float sMT[TILE_T];
    __shared__ float sMS[TILE_S];
    __shared__ float sRed[8];

    const int b  = blockIdx.z;
    const int t0 = blockIdx.y * TILE_T;
    const int s0 = blockIdx.x * TILE_S;

    const int tid  = threadIdx.x;
    const int lane = tid & 31;
    const int wave = tid >> 5;
    const int lm   = lane & 15;
    const int half = lane >> 4;

    const float* __restrict__ binp  = inp  + (size_t)b * kC * kT;
    const float* __restrict__ btgt  = tgt  + (size_t)b * kC * kT;
    const float* __restrict__ bmask = mask + (size_t)b * kT;

    if (tid < TILE_T) sMT[tid] = bmask[t0 + tid];
    if (tid < TILE_S) sMS[tid] = bmask[s0 + tid];

    // ---- per-thread staging geometry ----
    // t-unit: 2 K-rows x 4 t-cols of both x and y tiles.
    const int kpT = tid >> 5;            // 0..7
    const int tt4 = (tid & 31) << 2;     // 0..124
    // s-unit: threads <128 stage x s-tile, >=128 stage y s-tile.
    const int su   = tid & 127;
    const int kpS  = su >> 4;            // 0..7
    const int ss4  = (su & 15) << 2;     // 0..60
    const bool sIsY = tid >= 128;
    const float* __restrict__ sSrc = sIsY ? btgt : binp;

    auto load_t = [&](int c0) {
        const float* p = binp + (size_t)(c0 + 2 * kpT) * kT + t0 + tt4;
        const float* q = btgt + (size_t)(c0 + 2 * kpT) * kT + t0 + tt4;
        F4x4 r;
        r.x0 = *(const float4*)(p);
        r.x1 = *(const float4*)(p + kT);
        r.y0 = *(const float4*)(q);
        r.y1 = *(const float4*)(q + kT);
        return r;
    };
    auto load_s = [&](int c0) {
        const float* p = sSrc + (size_t)(c0 + 2 * kpS) * kT + s0 + ss4;
        F4x2 r;
        r.a0 = *(const float4*)(p);
        r.a1 = *(const float4*)(p + kT);
        return r;
    };
    auto store_t = [&](int buf, const F4x4& r) {
        const float m0 = sMT[tt4 + 0], m1 = sMT[tt4 + 1];
        const float m2 = sMT[tt4 + 2], m3 = sMT[tt4 + 3];
        const int base = (kpT * LD2T + tt4) * 2;
        float4 w;
        w.x = r.x0.x * m0; w.y = r.x1.x * m0; w.z = r.x0.y * m1; w.w = r.x1.y * m1;
        *(float4*)(&sXT[buf][base + 0]) = w;
        w.x = r.x0.z * m2; w.y = r.x1.z * m2; w.z = r.x0.w * m3; w.w = r.x1.w * m3;
        *(float4*)(&sXT[buf][base + 4]) = w;
        w.x = r.y0.x * m0; w.y = r.y1.x * m0; w.z = r.y0.y * m1; w.w = r.y1.y * m1;
        *(float4*)(&sYT[buf][base + 0]) = w;
        w.x = r.y0.z * m2; w.y = r.y1.z * m2; w.z = r.y0.w * m3; w.w = r.y1.w * m3;
        *(float4*)(&sYT[buf][base + 4]) = w;
    };
    auto store_s = [&](int buf, const F4x2& r) {
        const float m0 = sMS[ss4 + 0], m1 = sMS[ss4 + 1];
        const float m2 = sMS[ss4 + 2], m3 = sMS[ss4 + 3];
        float* dst = sIsY ? &sYS[buf][0] : &sXS[buf][0];
        const int base = (kpS * LD2S + ss4) * 2;
        float4 w;
        w.x = r.a0.x * m0; w.y = r.a1.x * m0; w.z = r.a0.y * m1; w.w = r.a1.y * m1;
        *(float4*)(dst + base + 0) = w;
        w.x = r.a0.z * m2; w.y = r.a1.z * m2; w.z = r.a0.w * m3; w.w = r.a1.w * m3;
        *(float4*)(dst + base + 4) = w;
    };

    // Per-wave 32x32 output region: 2x2 subtiles x {xy, xx, yy}
    v8f axy[2][2] = {};
    v8f axx[2][2] = {};
    v8f ayy[2][2] = {};

    const int wt = (wave & 3) * 32;    // 0,32,64,96
    const int ws = (wave >> 2) * 32;   // 0,32

    // Prefetch chunk 0 while masks land; barrier covers mask visibility.
    F4x4 pt = load_t(0);
    F4x2 ps = load_s(0);
    __syncthreads();

#pragma unroll 2
    for (int ci = 0; ci < NCHUNK; ++ci) {
        const int buf = ci & 1;
        // Drain prefetched chunk into LDS (mask applied here).
        store_t(buf, pt);
        store_s(buf, ps);
        // Fire next chunk's global loads; latency hides behind WMMA below.
        if (ci + 1 < NCHUNK) {
            pt = load_t((ci + 1) * KC);
            ps = load_s((ci + 1) * KC);
        }
        __syncthreads();

        // ---- WMMA over the chunk: K steps of 4, fragments via ds_load_b64 ----
#pragma unroll
        for (int k = 0; k < KC; k += 4) {
            const int kp = (k >> 1) + half;   // this half-wave's K pair row
            v2f ax[2], ay[2], bx[2], by[2];
#pragma unroll
            for (int i = 0; i < 2; ++i) {
                const int t = wt + i * 16 + lm;
                ax[i] = *(const v2f*)(&sXT[buf][(kp * LD2T + t) * 2]);
                ay[i] = *(const v2f*)(&sYT[buf][(kp * LD2T + t) * 2]);
                const int s = ws + i * 16 + lm;
                bx[i] = *(const v2f*)(&sXS[buf][(kp * LD2S + s) * 2]);
                by[i] = *(const v2f*)(&sYS[buf][(kp * LD2S + s) * 2]);
            }
#pragma unroll
            for (int mi = 0; mi < 2; ++mi)
#pragma unroll
                for (int ni = 0; ni < 2; ++ni) {
                    axy[mi][ni] = wmma_f32(ax[mi], by[ni], axy[mi][ni]);
                    axx[mi][ni] = wmma_f32(ax[mi], bx[ni], axx[mi][ni]);
                    ayy[mi][ni] = wmma_f32(ay[mi], by[ni], ayy[mi][ni]);
                }
        }
    }

    // ---- epilogue: loss = exp(-2*xy/(xx+yy)), reduce to scalar mean ----
    float partial = 0.0f;
#pragma unroll
    for (int mi = 0; mi < 2; ++mi)
#pragma unroll
        for (int ni = 0; ni < 2; ++ni)
#pragma unroll
            for (int e = 0; e < 8; ++e) {
                const float xy = axy[mi][ni][e];
                const float xx = axx[mi][ni][e];
                const float yy = ayy[mi][ni][e];
                partial += __expf(-2.0f * xy / (xx + yy));
            }

#pragma unroll
    for (int off = 16; off > 0; off >>= 1)
        partial += __shfl_xor(partial, off, 32);
    if (lane == 0) sRed[wave] = partial;
    __syncthreads();
    if (tid == 0) {
        float s = 0.0f;
#pragma unroll
        for (int i = 0; i < 8; ++i) s += sRed[i];
        atomicAdd(out, s * (1.0f / ((float)kB * (float)kT * (float)kT)));
    }
}

extern "C" void kernel_launch(void* const* d_in, const int* in_sizes, int n_in,
                              void* d_out, int out_size, void* d_ws, size_t ws_size,
                              hipStream_t stream) {
    const float* inp  = (const float*)d_in[0];
    const float* tgt  = (const float*)d_in[1];
    const float* mask = (const float*)d_in[2];
    float* out = (float*)d_out;

    jvs_zero<<<1, 32, 0, stream>>>(out);
    dim3 grid(kT / TILE_S, kT / TILE_T, kB);  // (16, 8, 16)
    jvs_main<<<grid, 256, 0, stream>>>(inp, tgt, mask, out);
}